// AssociativeMemory_88381837017745
// MI455X (gfx1250) — compile-verified
//
#include <hip/hip_runtime.h>

#define B_   2048
#define S_   128
#define D_   256
#define TWO_D 512

typedef __attribute__((ext_vector_type(2))) float v2f;
typedef __attribute__((ext_vector_type(8))) float v8f;

__device__ __forceinline__ float wave_sum(float v) {
    // wave32 butterfly reduction
    for (int i = 16; i > 0; i >>= 1) v += __shfl_xor(v, i, 32);
    return v;
}

// --- block-wide tree reductions through LDS (all 256 threads must call) ---
__device__ __forceinline__ float block_max128(float* red, int tid, float v) {
    if (tid < 128) red[tid] = v;
    __syncthreads();
    for (int st = 64; st > 0; st >>= 1) {
        if (tid < st) red[tid] = fmaxf(red[tid], red[tid + st]);
        __syncthreads();
    }
    float r = red[0];
    __syncthreads();
    return r;
}
__device__ __forceinline__ float block_sum128(float* red, int tid, float v) {
    if (tid < 128) red[tid] = v;
    __syncthreads();
    for (int st = 64; st > 0; st >>= 1) {
        if (tid < st) red[tid] += red[tid + st];
        __syncthreads();
    }
    float r = red[0];
    __syncthreads();
    return r;
}
__device__ __forceinline__ float block_sum256(float* red, int tid, float v) {
    red[tid] = v;
    __syncthreads();
    for (int st = 128; st > 0; st >>= 1) {
        if (tid < st) red[tid] += red[tid + st];
        __syncthreads();
    }
    float r = red[0];
    __syncthreads();
    return r;
}

// CDNA5 async global->LDS copy (ASYNCcnt tracked, bypasses VGPRs).
// ISA 10.2: flat pointer to LDS has LDS byte offset in addr[31:0].
__device__ __forceinline__ void async_load_lds_b128(void* lds_dst, const void* gsrc) {
    unsigned           lds_off = (unsigned)(unsigned long long)lds_dst;
    unsigned long long gaddr   = (unsigned long long)gsrc;
    asm volatile("global_load_async_to_lds_b128 %0, %1, off"
                 :: "v"(lds_off), "v"(gaddr) : "memory");
}
__device__ __forceinline__ void wait_asynccnt0() {
    asm volatile("s_wait_asynccnt 0x0" ::: "memory");
}

__global__ void zero_scalar_kernel(float* p) { *p = 0.0f; }

// logits[b,s] = concat(gw_real[b], gw_imag[b]) . Wa[:,s] + ba[s]
// One wave computes one 16(M=batch) x 16(N=slot) tile with V_WMMA_F32_16X16X4_F32.
__global__ __launch_bounds__(32)
void logits_wmma_kernel(const float* __restrict__ gwr,
                        const float* __restrict__ gwi,
                        const float* __restrict__ Wa,   // [512,128]
                        const float* __restrict__ ba,   // [128]
                        float* __restrict__ logits)     // [B,S]
{
    const int lane  = threadIdx.x;        // 0..31
    const int m0    = blockIdx.x * 16;    // batch tile
    const int n0    = blockIdx.y * 16;    // slot tile
    const int mrow  = m0 + (lane & 15);   // A row for this lane
    const int ncol  = n0 + (lane & 15);   // B col for this lane
    const int khalf = (lane >> 4) * 2;    // 0 for lanes 0-15, 2 for lanes 16-31

    v8f c = {0.f, 0.f, 0.f, 0.f, 0.f, 0.f, 0.f, 0.f};

    for (int k0 = 0; k0 < TWO_D; k0 += 4) {
        const int ka = k0 + khalf;
        v2f a, b;
        // A (16x4, f32): VGPR0 = K=khalf, VGPR1 = K=khalf+1 (ISA 7.12.2)
        a.x = (ka     < D_) ? gwr[(size_t)mrow * D_ + ka]     : gwi[(size_t)mrow * D_ + (ka - D_)];
        a.y = (ka + 1 < D_) ? gwr[(size_t)mrow * D_ + ka + 1] : gwi[(size_t)mrow * D_ + (ka + 1 - D_)];
        // B (4x16, f32): row striped across lanes, K rows split like A
        b.x = Wa[(size_t)(ka)     * S_ + ncol];
        b.y = Wa[(size_t)(ka + 1) * S_ + ncol];
        c = __builtin_amdgcn_wmma_f32_16x16x4_f32(false, a, false, b,
                                                  (short)0, c, false, false);
    }

    // C/D layout: VGPR j -> M=j (lanes 0-15) / M=j+8 (lanes 16-31), N=lane&15
    const int   mbase = m0 + ((lane >> 4) * 8);
    const float bav   = ba[ncol];
    for (int j = 0; j < 8; ++j) {
        logits[(size_t)(mbase + j) * S_ + ncol] = c[j] + bav;
    }
}

// One workgroup (256 threads = 8 waves) per batch element.
__global__ __launch_bounds__(256)
void assoc_mem_kernel(const float* __restrict__ gwr_g,
                      const float* __restrict__ gwi_g,
                      const float* __restrict__ memr_g,
                      const float* __restrict__ memi_g,
                      const float* __restrict__ Wg,
                      const float* __restrict__ bg,
                      const float* __restrict__ gamma_r,
                      const float* __restrict__ beta_r,
                      const float* __restrict__ gamma_i,
                      const float* __restrict__ beta_i,
                      const float* __restrict__ logits_ws,
                      float* __restrict__ read_r,
                      float* __restrict__ read_i,
                      float* __restrict__ next_r,
                      float* __restrict__ next_i,
                      float* __restrict__ entropy_out)
{
    const int b    = blockIdx.x;
    const int tid  = threadIdx.x;
    const int lane = tid & 31;
    const int wave = tid >> 5;   // 0..7

    extern __shared__ float smem[];           // 2 * S_ * D_ floats (256 KB)
    float* s_real = smem;
    float* s_imag = smem + S_ * D_;

    __shared__ float gwr[D_], gwi[D_];
    __shared__ float gmr[D_], gbr[D_], gmi[D_], gbi[D_];
    __shared__ float attn[S_];
    __shared__ float esh[S_];
    __shared__ float effv[S_];
    __shared__ float red[256];
    __shared__ float s_gate, s_invsum, s_invdenom;
    __shared__ int   s_top[3];

    // ---- stage per-batch vectors + LN params (normal DS stores) ----
    gwr[tid] = gwr_g[(size_t)b * D_ + tid];
    gwi[tid] = gwi_g[(size_t)b * D_ + tid];
    gmr[tid] = gamma_r[tid]; gbr[tid] = beta_r[tid];
    gmi[tid] = gamma_i[tid]; gbi[tid] = beta_i[tid];

    // ---- kick off async staging of mem[b] into LDS (ASYNCcnt path) ----
    {
        const float4* mr4 = (const float4*)(memr_g + (size_t)b * S_ * D_);
        const float4* mi4 = (const float4*)(memi_g + (size_t)b * S_ * D_);
        float4* sr4 = (float4*)s_real;
        float4* si4 = (float4*)s_imag;
        for (int i = tid; i < (S_ * D_) / 4; i += 256) {
            async_load_lds_b128(sr4 + i, mr4 + i);
            async_load_lds_b128(si4 + i, mi4 + i);
        }
    }
    __syncthreads();   // gwr/gwi visible; async copies still in flight

    // =====================================================================
    // Work that does NOT need mem[b] — overlapped with the async LDS fill.
    // =====================================================================

    // ---- write gate: sigmoid(flat . Wg + bg) ----
    {
        float g = block_sum256(red, tid, gwr[tid] * Wg[tid] + gwi[tid] * Wg[D_ + tid]);
        if (tid == 0) s_gate = 1.f / (1.f + expf(-(g + bg[0])));
    }

    // ---- softmax(logits), entropy, top-3 sparse write weights ----
    {
        float lg = (tid < S_) ? logits_ws[(size_t)b * S_ + tid] : -1e30f;
        float mx = block_max128(red, tid, lg);
        float e  = (tid < S_) ? expf(lg - mx) : 0.f;
        if (tid < S_) esh[tid] = e;
        float sm  = block_sum128(red, tid, e);
        float inv = 1.f / sm;
        float w   = e * inv;
        float ent = block_sum128(red, tid, (tid < S_) ? (-w * logf(w + 1e-10f)) : 0.f);
        if (tid == 0) {
            atomicAdd(entropy_out, ent * (1.f / (float)B_));
            // top-3 (strict '>' keeps first index on ties, matches jax top_k)
            int   i0 = -1, i1 = -1, i2 = -1;
            float v0 = -1e30f, v1 = -1e30f, v2 = -1e30f;
            for (int s = 0; s < S_; ++s) {
                const float v = esh[s];
                if (v > v0)      { v2 = v1; i2 = i1; v1 = v0; i1 = i0; v0 = v; i0 = s; }
                else if (v > v1) { v2 = v1; i2 = i1; v1 = v; i1 = s; }
                else if (v > v2) { v2 = v; i2 = s; }
            }
            s_top[0] = i0; s_top[1] = i1; s_top[2] = i2;
            const float topw = (v0 + v1 + v2) * inv;
            s_invsum   = inv;
            s_invdenom = 1.f / (topw + 1e-6f);
        }
        __syncthreads();
        if (tid < S_) {
            const bool in3 = (tid == s_top[0]) | (tid == s_top[1]) | (tid == s_top[2]);
            effv[tid] = in3 ? (s_gate * (esh[tid] * s_invsum) * s_invdenom) : 0.f;
        }
    }

    // =====================================================================
    // mem[b] needed from here on: drain async copies, then barrier.
    // =====================================================================
    wait_asynccnt0();
    __syncthreads();

    // ---- sim[s] = mem_real[s,:].gw_real + mem_imag[s,:].gw_imag ----
    for (int s = wave; s < S_; s += 8) {
        const float* rr = s_real + s * D_;
        const float* ri = s_imag + s * D_;
        float p = 0.f;
        for (int j = 0; j < 8; ++j) {
            int d = lane + 32 * j;                  // conflict-free LDS pattern
            p += rr[d] * gwr[d] + ri[d] * gwi[d];
        }
        p = wave_sum(p);
        if (lane == 0) esh[s] = p;                  // esh now holds sim
    }
    __syncthreads();

    // ---- softmax(sim) -> attn ----
    {
        float sv = (tid < S_) ? esh[tid] : -1e30f;
        float mx = block_max128(red, tid, sv);
        float e  = (tid < S_) ? expf(sv - mx) : 0.f;
        float sm = block_sum128(red, tid, e);
        if (tid < S_) attn[tid] = e / sm;
    }
    __syncthreads();

    // ---- read_r / read_i (each thread owns one d) ----
    {
        float ar = 0.f, ai = 0.f;
        for (int s = 0; s < S_; ++s) {
            const float a = attn[s];
            ar += a * s_real[s * D_ + tid];
            ai += a * s_imag[s * D_ + tid];
        }
        read_r[(size_t)b * D_ + tid] = ar;
        read_i[(size_t)b * D_ + tid] = ai;
    }

    // ---- next = LayerNorm(tanh(mem + eff*(gw - mem))) ----
    const float invD = 1.f / (float)D_;
    for (int s = wave; s < S_; s += 8) {
        const float* rr = s_real + s * D_;
        const float* ri = s_imag + s * D_;
        const float e = effv[s];
        float tr[8], ti[8];
        float sumr = 0.f, sumi = 0.f, sqr = 0.f, sqi = 0.f;
        for (int j = 0; j < 8; ++j) {
            const int d = lane + 32 * j;
            const float mrv = rr[d], miv = ri[d];
            const float vr = tanhf(mrv + e * (gwr[d] - mrv));
            const float vi = tanhf(miv + e * (gwi[d] - miv));
            tr[j] = vr; ti[j] = vi;
            sumr += vr; sumi += vi;
            sqr += vr * vr; sqi += vi * vi;
        }
        sumr = wave_sum(sumr); sumi = wave_sum(sumi);
        sqr  = wave_sum(sqr);  sqi  = wave_sum(sqi);
        const float mr  = sumr * invD;
        const float mi  = sumi * invD;
        const float isr = rsqrtf(sqr * invD - mr * mr + 1e-6f);
        const float isi = rsqrtf(sqi * invD - mi * mi + 1e-6f);
        const size_t base = ((size_t)b * S_ + s) * D_;
        for (int j = 0; j < 8; ++j) {
            const int d = lane + 32 * j;            // coalesced across the wave
            next_r[base + d] = (tr[j] - mr) * isr * gmr[d] + gbr[d];
            next_i[base + d] = (ti[j] - mi) * isi * gmi[d] + gbi[d];
        }
    }
}

extern "C" void kernel_launch(void* const* d_in, const int* in_sizes, int n_in,
                              void* d_out, int out_size, void* d_ws, size_t ws_size,
                              hipStream_t stream)
{
    const float* gwr  = (const float*)d_in[0];
    const float* gwi  = (const float*)d_in[1];
    const float* memr = (const float*)d_in[2];
    const float* memi = (const float*)d_in[3];
    const float* Wg   = (const float*)d_in[4];
    const float* bg   = (const float*)d_in[5];
    const float* Wa   = (const float*)d_in[6];
    const float* ba   = (const float*)d_in[7];
    const float* gr   = (const float*)d_in[8];
    const float* br   = (const float*)d_in[9];
    const float* gi   = (const float*)d_in[10];
    const float* bi   = (const float*)d_in[11];

    float* out      = (float*)d_out;
    float* read_r   = out;
    float* read_i   = out + (size_t)B_ * D_;
    float* next_r   = out + 2 * (size_t)B_ * D_;
    float* next_i   = next_r + (size_t)B_ * S_ * D_;
    float* entropy  = next_i + (size_t)B_ * S_ * D_;

    float* logits = (float*)d_ws;   // B_*S_ floats = 1 MB

    zero_scalar_kernel<<<1, 1, 0, stream>>>(entropy);

    dim3 gg(B_ / 16, S_ / 16);      // 128 x 8 WMMA tiles
    logits_wmma_kernel<<<gg, 32, 0, stream>>>(gwr, gwi, Wa, ba, logits);

    const size_t lds_bytes = (size_t)2 * S_ * D_ * sizeof(float);  // 256 KB of the 320 KB WGP LDS
    assoc_mem_kernel<<<B_, 256, lds_bytes, stream>>>(
        gwr, gwi, memr, memi, Wg, bg, gr, br, gi, bi, logits,
        read_r, read_i, next_r, next_i, entropy);
}